// GCN_RW_mini_G_13975823581635
// MI455X (gfx1250) — compile-verified
//
#include <hip/hip_runtime.h>
#include <math.h>

typedef __attribute__((ext_vector_type(2))) float v2f;
typedef __attribute__((ext_vector_type(8))) float v8f;

#define NNODES 200000
#define BATCH  8192
#define RWS    8
#define KHOPS  3
#define HDIM   256

#define BM 128          // rows per block (8 waves x 16)
#define BN 64           // cols per block
#define KT 32           // k-chunk staged in LDS

// ---------------------------------------------------------------------------
// map[i] = position of node i in batch, or -1 if not in batch
// ---------------------------------------------------------------------------
__global__ void init_map_kernel(int* __restrict__ map_, int n) {
    int i = blockIdx.x * blockDim.x + threadIdx.x;
    if (i < n) map_[i] = -1;
}
__global__ void set_map_kernel(const int* __restrict__ batch, int* __restrict__ map_, int bn) {
    int j = blockIdx.x * blockDim.x + threadIdx.x;
    if (j < bn) map_[batch[j]] = j;
}

// ---------------------------------------------------------------------------
// Y[M x Hd] = X[M x Kd] @ W[Kd x Hd] + bias   (fp32, V_WMMA_F32_16X16X4_F32)
//
// Block (256 thr / 8 waves) computes a 128x64 tile. The 64-wide B panel is
// shared by all 8 waves through a double-buffered LDS stage (KT x BN fp32,
// 2 x 8KB), loaded cooperatively & coalesced -> 8x less B traffic than the
// wave-private scheme. Wave w handles rows [bm + 16w, +16), all 64 cols
// (4 accumulators, A fragment reused across the 4 WMMAs per k-step).
//
// A frag (16x4 f32, ISA 7.12.2): lane = 16*h + m  (h = lane>>4, m = lane&15)
//   vgpr0 = A[m][k + 2h + 0], vgpr1 = A[m][k + 2h + 1]
// B frag (4x16): rows striped across lanes within a VGPR:
//   vgpr0 = B[k + 2h + 0][n], vgpr1 = B[k + 2h + 1][n], n = lane&15
// C/D: vgpr r = C[8h + r][n]
// Grid is exact (no bounds checks) -> EXEC all-1s at every WMMA.
// ---------------------------------------------------------------------------
__global__ __launch_bounds__(256) void gemm_f32_wmma_kernel(
    const float* __restrict__ X, const float* __restrict__ W,
    const float* __restrict__ bias, float* __restrict__ Y,
    int Kd, int Hd)
{
    __shared__ float sB[2][KT][BN];     // 16 KB

    const int nblk = Hd / BN;
    const int bm   = (blockIdx.x / nblk) * BM;
    const int bn   = (blockIdx.x % nblk) * BN;

    const int t    = threadIdx.x;
    const int w    = t >> 5;
    const int lane = t & 31;
    const int m15  = lane & 15;
    const int half = lane >> 4;         // 0 or 1
    const int khalf = half * 2;         // 0 or 2

    const float* xrow = X + (size_t)(bm + w * 16 + m15) * Kd + khalf;
    const float* wsrc = W + bn + (t & 63);        // + kk*Hd
    const int    ldk  = t >> 6;                   // base k-row this thread loads
    const int    ldn  = t & 63;

    const int nchunks = Kd / KT;

    // prologue: stage chunk 0
    {
        float r[8];
#pragma unroll
        for (int i = 0; i < 8; ++i)
            r[i] = wsrc[(size_t)(ldk + i * 4) * Hd];
#pragma unroll
        for (int i = 0; i < 8; ++i)
            sB[0][ldk + i * 4][ldn] = r[i];
    }
    __syncthreads();

    v8f acc0 = {}, acc1 = {}, acc2 = {}, acc3 = {};

    for (int c = 0; c < nchunks; ++c) {
        const int buf = c & 1;
        float r[8];
        const bool more = (c + 1 < nchunks);
        if (more) {
            const float* wnext = wsrc + (size_t)(c + 1) * KT * Hd;
#pragma unroll
            for (int i = 0; i < 8; ++i)
                r[i] = wnext[(size_t)(ldk + i * 4) * Hd];
            __builtin_prefetch(xrow + (c + 1) * KT, 0, 3);   // global_prefetch_b8
        }

        const float* xk = xrow + c * KT;
#pragma unroll
        for (int kk = 0; kk < KT; kk += 4) {
            v2f a;
            a.x = xk[kk + 0];
            a.y = xk[kk + 1];
            const float* b_lo = &sB[buf][kk + khalf + 0][m15];
            const float* b_hi = &sB[buf][kk + khalf + 1][m15];
            v2f b0, b1, b2, b3;
            b0.x = b_lo[0];  b0.y = b_hi[0];
            b1.x = b_lo[16]; b1.y = b_hi[16];
            b2.x = b_lo[32]; b2.y = b_hi[32];
            b3.x = b_lo[48]; b3.y = b_hi[48];
            acc0 = __builtin_amdgcn_wmma_f32_16x16x4_f32(false, a, false, b0, (short)0, acc0, false, false);
            acc1 = __builtin_amdgcn_wmma_f32_16x16x4_f32(false, a, false, b1, (short)0, acc1, false, false);
            acc2 = __builtin_amdgcn_wmma_f32_16x16x4_f32(false, a, false, b2, (short)0, acc2, false, false);
            acc3 = __builtin_amdgcn_wmma_f32_16x16x4_f32(false, a, false, b3, (short)0, acc3, false, false);
        }

        if (more) {
#pragma unroll
            for (int i = 0; i < 8; ++i)
                sB[buf ^ 1][ldk + i * 4][ldn] = r[i];
        }
        __syncthreads();
    }

    // writeback: 4 subtiles; rows bm + 16w + 8*half + r, cols bn + 16j + m15
    const int rowbase = bm + w * 16 + half * 8;
    v8f accs[4] = {acc0, acc1, acc2, acc3};
#pragma unroll
    for (int j = 0; j < 4; ++j) {
        const int   cn = bn + j * 16 + m15;
        const float bv = bias[cn];
        float* yout = Y + (size_t)rowbase * Hd + cn;
#pragma unroll
        for (int rr = 0; rr < 8; ++rr)
            yout[(size_t)rr * Hd] = accs[j][rr] + bv;
    }
}

// ---------------------------------------------------------------------------
// Random-walk aggregation + ReLU for one layer.
// One block (HDIM threads) per batch row; thread t handles feature t.
// Walk indices / degrees are block-uniform -> L1-hot; row gathers coalesced.
// ---------------------------------------------------------------------------
__global__ __launch_bounds__(HDIM) void aggr_relu_kernel(
    const float* __restrict__ Yin,    // [BATCH x HDIM] fresh layer output
    const float* __restrict__ hist,   // [NNODES x HDIM] this layer's history
    const float* __restrict__ degree, // [NNODES]
    const float* __restrict__ att,    // [KHOPS+1] this layer's mask
    const int*   __restrict__ ends,   // [KHOPS x BATCH*RWS] this layer
    const int*   __restrict__ map_,   // [NNODES]
    const int*   __restrict__ batch,  // [BATCH]
    float*       __restrict__ Xout)   // [BATCH x HDIM]
{
    const int i = blockIdx.x;
    const int t = threadIdx.x;

    float acc = Yin[(size_t)i * HDIM + t] * att[0];
    const float dstart = sqrtf(degree[batch[i]]);

#pragma unroll
    for (int k = 1; k <= KHOPS; ++k) {
        const float mk = att[k] * (1.0f / (float)RWS);
        const int*  ek = ends + (size_t)(k - 1) * (BATCH * RWS) + i * RWS;
        float sum = 0.0f;
#pragma unroll
        for (int r = 0; r < RWS; ++r) {
            const int   e = ek[r];
            const float w = dstart * rsqrtf(degree[e]);
            const int   m = map_[e];
            const float* src = (m >= 0) ? (Yin  + (size_t)m * HDIM)
                                        : (hist + (size_t)e * HDIM);
            sum += w * src[t];
        }
        acc += sum * mk;
    }
    Xout[(size_t)i * HDIM + t] = fmaxf(acc, 0.0f);
}

// ---------------------------------------------------------------------------
// Output head: logits = X2 @ Wout + bout, then row-wise log_softmax.
// One block (HDIM threads) per row; C=47 classes.
// ---------------------------------------------------------------------------
__global__ __launch_bounds__(HDIM) void head_kernel(
    const float* __restrict__ X2,   // [BATCH x HDIM]
    const float* __restrict__ Wout, // [HDIM x C]
    const float* __restrict__ bout, // [C]
    float*       __restrict__ Out,  // [BATCH x C]
    int Cc)
{
    __shared__ float sx[HDIM];
    __shared__ float slogit[64];
    __shared__ float smax, slse;

    const int i = blockIdx.x;
    const int t = threadIdx.x;

    sx[t] = X2[(size_t)i * HDIM + t];
    __syncthreads();

    if (t < Cc) {
        float a = bout[t];
        for (int h = 0; h < HDIM; ++h)
            a = fmaf(sx[h], Wout[(size_t)h * Cc + t], a);
        slogit[t] = a;
    }
    __syncthreads();

    if (t == 0) {
        float mx = -INFINITY;
        for (int c = 0; c < Cc; ++c) mx = fmaxf(mx, slogit[c]);
        float s = 0.0f;
        for (int c = 0; c < Cc; ++c) s += expf(slogit[c] - mx);
        smax = mx;
        slse = logf(s);
    }
    __syncthreads();

    if (t < Cc)
        Out[(size_t)i * Cc + t] = slogit[t] - smax - slse;
}

// ---------------------------------------------------------------------------
// Host-side launch
// ---------------------------------------------------------------------------
extern "C" void kernel_launch(void* const* d_in, const int* in_sizes, int n_in,
                              void* d_out, int out_size, void* d_ws, size_t ws_size,
                              hipStream_t stream)
{
    const float* x_feat  = (const float*)d_in[0];   // [B, 512]
    const float* histEmb = (const float*)d_in[1];   // [2, N, H]
    const float* degree  = (const float*)d_in[2];   // [N]
    const float* att     = (const float*)d_in[3];   // [2, 4]
    const float* W0      = (const float*)d_in[4];   // [512, 256]
    const float* b0      = (const float*)d_in[5];   // [256]
    const float* W1      = (const float*)d_in[6];   // [256, 256]
    const float* b1      = (const float*)d_in[7];   // [256]
    const float* Wout    = (const float*)d_in[8];   // [256, 47]
    const float* bout    = (const float*)d_in[9];   // [47]
    const int*   batch   = (const int*)d_in[10];    // [B]
    const int*   ends    = (const int*)d_in[11];    // [2, 3, B*RWS]

    const int F_IN = 512;
    const int C    = in_sizes[9];                   // 47

    // workspace layout
    char* ws   = (char*)d_ws;
    int*  map_ = (int*)ws;                                        // N ints
    size_t off = ((size_t)NNODES * sizeof(int) + 255) & ~(size_t)255;
    float* buf0 = (float*)(ws + off);                             // [B x H]
    float* buf1 = (float*)(ws + off + (size_t)BATCH * HDIM * sizeof(float));

    // build in-batch map
    init_map_kernel<<<(NNODES + 255) / 256, 256, 0, stream>>>(map_, NNODES);
    set_map_kernel<<<(BATCH + 255) / 256, 256, 0, stream>>>(batch, map_, BATCH);

    // GEMM grid: 128x64 tile per 256-thread block
    const int gblocks = (BATCH / BM) * (HDIM / BN);   // 64 * 4 = 256

    // ----- layer 0 -----
    gemm_f32_wmma_kernel<<<gblocks, 256, 0, stream>>>(
        x_feat, W0, b0, buf0, F_IN, HDIM);
    aggr_relu_kernel<<<BATCH, HDIM, 0, stream>>>(
        buf0, histEmb, degree, att, ends, map_, batch, buf1);

    // ----- layer 1 -----
    gemm_f32_wmma_kernel<<<gblocks, 256, 0, stream>>>(
        buf1, W1, b1, buf0, HDIM, HDIM);
    aggr_relu_kernel<<<BATCH, HDIM, 0, stream>>>(
        buf0, histEmb + (size_t)NNODES * HDIM, degree, att + (KHOPS + 1),
        ends + (size_t)KHOPS * BATCH * RWS, map_, batch, buf1);

    // ----- head -----
    head_kernel<<<BATCH, HDIM, 0, stream>>>(buf1, Wout, bout, (float*)d_out, C);
}